// KGMC_77919296684698
// MI455X (gfx1250) — compile-verified
//
#include <hip/hip_runtime.h>
#include <math.h>

typedef __attribute__((ext_vector_type(2))) float v2f;
typedef __attribute__((ext_vector_type(4))) float v4f;
typedef __attribute__((ext_vector_type(8))) float v8f;

#define N_NODES 200000
#define N_EDGES 1600000
#define N_PAIRS 4096
#define NB      2
#define NREL    8
#define NTILES  (N_NODES / 16)   // 12500, exact

// ---------------------------------------------------------------------------
// Node transform: for a 16-row tile of nodes compute
//   hb[n, b, :]  = h[n, :di] @ basis[b]           (b = 0,1)
//   agg[n, :]    = h[n, :di] @ wself + bias       (seed for edge scatter-add)
// with V_WMMA_F32_16X16X4_F32.  One wave per tile, 8 waves per block.
//
// LDS weight layout is K-pair interleaved:  element (mat j, k, col) lives at
//   ldsW[j*DI*32 + (k>>1)*64 + col*2 + (k&1)]
// so a B fragment {W[k0][c], W[k0+1][c]} is ONE aligned ds_load_b64 straight
// into the WMMA operand register pair (no v_mov repacking).
// ---------------------------------------------------------------------------
template <int DI>
__global__ __launch_bounds__(256) void node_transform_kernel(
    const float* __restrict__ h, int ldh, int hoff,
    const float* __restrict__ basis,   // [NB, DI, 32]
    const float* __restrict__ wself,   // [DI, 32]
    const float* __restrict__ bias,    // [32]
    float* __restrict__ hb,            // [N, NB*32]
    float* __restrict__ agg)           // [N, 32]
{
    __shared__ float ldsW[3 * DI * 32];

    const int tid = threadIdx.x;
    // cooperative staging with pair-interleave scatter
    for (int i = tid; i < 3 * DI * 32; i += 256) {
        const int j = i / (DI * 32);
        const int r = i - j * DI * 32;
        const int k = r >> 5;            // r / 32
        const int c = r & 31;
        const float val = (j < 2) ? basis[i] : wself[r];
        ldsW[j * DI * 32 + (k >> 1) * 64 + c * 2 + (k & 1)] = val;
    }
    __syncthreads();

    const int lane  = tid & 31;
    const int row16 = lane & 15;      // M index (A) / N index (B,C,D)
    const int half  = lane >> 4;      // selects K-pair (A/B) / M+8 (C,D)
    const int tile  = blockIdx.x * 8 + (tid >> 5);
    if (tile * 16 >= N_NODES) return;

    v8f acc[6];                       // [mat(3)][ntile(2)]
#pragma unroll
    for (int i = 0; i < 6; ++i) acc[i] = (v8f){0.f, 0.f, 0.f, 0.f, 0.f, 0.f, 0.f, 0.f};

    // A fragment: lane holds A[row16][k0], A[row16][k0+1], k0 = ks*4 + half*2
    const float* arow = h + (size_t)(tile * 16 + row16) * ldh + hoff + half * 2;

#pragma unroll
    for (int ks = 0; ks < DI / 4; ++ks) {
        const v2f a = *(const v2f*)(arow + ks * 4);          // 8B aligned
        const int kp = ks * 2 + half;                         // k0 >> 1
#pragma unroll
        for (int j = 0; j < 3; ++j) {
#pragma unroll
            for (int nt = 0; nt < 2; ++nt) {
                const v2f b = *(const v2f*)&ldsW[j * DI * 32 + kp * 64
                                                 + (nt * 16 + row16) * 2];
                acc[j * 2 + nt] = __builtin_amdgcn_wmma_f32_16x16x4_f32(
                    false, a, false, b, (short)0, acc[j * 2 + nt], false, false);
            }
        }
    }

    const float biasv0 = bias[row16];
    const float biasv1 = bias[16 + row16];

    // C/D layout: VGPR jv -> M = jv + half*8, N = row16 (+ nt*16)
#pragma unroll
    for (int jv = 0; jv < 8; ++jv) {
        const size_t grow = (size_t)(tile * 16 + half * 8 + jv);
#pragma unroll
        for (int nt = 0; nt < 2; ++nt) {
            const int col = nt * 16 + row16;
            hb[grow * 64 + col]      = acc[0 * 2 + nt][jv];          // basis 0
            hb[grow * 64 + 32 + col] = acc[1 * 2 + nt][jv];          // basis 1
            agg[grow * 32 + col]     = acc[2 * 2 + nt][jv] + (nt ? biasv1 : biasv0);
        }
    }
}

// ---------------------------------------------------------------------------
// Edge phase: wave cooperatively processes 32 edges; lane = output feature.
// Edge indices are loaded once per lane (coalesced) and broadcast via __shfl;
// the 16-entry comp table lives in wave registers.  Hot loop per edge:
// 2 coalesced 128B gathers + 1 global_atomic_add_f32 (all L2-resident).
// ---------------------------------------------------------------------------
__global__ __launch_bounds__(256) void edge_kernel(
    const int* __restrict__ src, const int* __restrict__ dst,
    const int* __restrict__ etype, const float* __restrict__ comp, // [R, NB]
    const float* __restrict__ hb, float* __restrict__ agg)
{
    const int lane   = threadIdx.x & 31;
    const int gwave  = (blockIdx.x * blockDim.x + threadIdx.x) >> 5;
    const int nwaves = (gridDim.x * blockDim.x) >> 5;

    const float cw = (lane < NREL * NB) ? comp[lane] : 0.f;

    for (int base = gwave * 32; base < N_EDGES; base += nwaves * 32) {
        const int e = base + lane;
        int sl = 0, dl = 0, tl = 0;
        if (e < N_EDGES) { sl = src[e]; dl = dst[e]; tl = etype[e]; }
        const int rem = N_EDGES - base;
        const int cnt = rem < 32 ? rem : 32;
        for (int i = 0; i < cnt; ++i) {
            const int   s  = __shfl(sl, i, 32);
            const int   d  = __shfl(dl, i, 32);
            const int   t  = __shfl(tl, i, 32);
            const float w0 = __shfl(cw, t * NB + 0, 32);
            const float w1 = __shfl(cw, t * NB + 1, 32);
            const float m  = w0 * hb[(size_t)s * 64 + lane]
                           + w1 * hb[(size_t)s * 64 + 32 + lane];
            atomicAdd(&agg[(size_t)d * 32 + lane], m);
        }
    }
}

// ---------------------------------------------------------------------------
// tanh + write layer state into concat buffer cat[N,128] (column block l*32).
// float4 vectorized: 128-bit loads/stores.
// ---------------------------------------------------------------------------
__global__ __launch_bounds__(256) void tanh_cat_kernel(
    const float* __restrict__ agg, float* __restrict__ cat, int layer)
{
    const int total  = N_NODES * 8;            // groups of 4 features
    const int stride = gridDim.x * blockDim.x;
    for (int i = blockIdx.x * blockDim.x + threadIdx.x; i < total; i += stride) {
        const int n  = i >> 3;
        const int og = (i & 7) * 4;
        v4f v = *(const v4f*)&agg[(size_t)n * 32 + og];
        v4f r;
        r.x = tanhf(v.x); r.y = tanhf(v.y); r.z = tanhf(v.z); r.w = tanhf(v.w);
        *(v4f*)&cat[(size_t)n * 128 + layer * 32 + og] = r;
    }
}

// ---------------------------------------------------------------------------
// Pair scorer: wave per pair; 256-wide dot (cat[u]||cat[i]) . w, sigmoid.
// ---------------------------------------------------------------------------
__global__ __launch_bounds__(256) void pair_kernel(
    const int* __restrict__ uidx, const int* __restrict__ iidx,
    const float* __restrict__ cat, const float* __restrict__ w,  // [256]
    const float* __restrict__ b, float* __restrict__ out)
{
    const int lane = threadIdx.x & 31;
    const int p    = blockIdx.x * 8 + (threadIdx.x >> 5);
    if (p >= N_PAIRS) return;

    const int u  = uidx[p];
    const int it = iidx[p];
    float acc = 0.f;
#pragma unroll
    for (int j = 0; j < 4; ++j) {
        const int c = j * 32 + lane;
        acc += cat[(size_t)u  * 128 + c] * w[c];
        acc += cat[(size_t)it * 128 + c] * w[128 + c];
    }
#pragma unroll
    for (int off = 16; off > 0; off >>= 1)
        acc += __shfl_down(acc, off, 32);
    if (lane == 0)
        out[p] = 1.f / (1.f + expf(-(acc + b[0])));
}

// ---------------------------------------------------------------------------
extern "C" void kernel_launch(void* const* d_in, const int* in_sizes, int n_in,
                              void* d_out, int out_size, void* d_ws, size_t ws_size,
                              hipStream_t stream)
{
    const float* x     = (const float*)d_in[0];
    const int*   src   = (const int*)d_in[1];
    const int*   dst   = (const int*)d_in[2];
    const int*   etype = (const int*)d_in[3];
    const int*   uidx  = (const int*)d_in[4];
    const int*   iidx  = (const int*)d_in[5];

    const float* basis[4], *comp[4], *wself[4], *bias[4];
    for (int l = 0; l < 4; ++l) {
        basis[l] = (const float*)d_in[6 + 4 * l + 0];
        comp[l]  = (const float*)d_in[6 + 4 * l + 1];
        wself[l] = (const float*)d_in[6 + 4 * l + 2];
        bias[l]  = (const float*)d_in[6 + 4 * l + 3];
    }
    const float* lin1w = (const float*)d_in[22];
    const float* lin1b = (const float*)d_in[23];

    float* ws  = (float*)d_ws;
    float* hb  = ws;                                   // N*64 f32  (51.2 MB)
    float* agg = hb + (size_t)N_NODES * 64;            // N*32 f32  (25.6 MB)
    float* cat = agg + (size_t)N_NODES * 32;           // N*128 f32 (102.4 MB)

    const int ntBlocks = (NTILES + 7) / 8;             // 8 waves/block

    for (int l = 0; l < 4; ++l) {
        const float* hsrc = (l == 0) ? x : cat;
        const int ldh  = (l == 0) ? 64 : 128;
        const int hoff = (l == 0) ? 0 : (l - 1) * 32;
        if (l == 0)
            node_transform_kernel<64><<<ntBlocks, 256, 0, stream>>>(
                hsrc, ldh, hoff, basis[l], wself[l], bias[l], hb, agg);
        else
            node_transform_kernel<32><<<ntBlocks, 256, 0, stream>>>(
                hsrc, ldh, hoff, basis[l], wself[l], bias[l], hb, agg);

        edge_kernel<<<1024, 256, 0, stream>>>(src, dst, etype, comp[l], hb, agg);
        tanh_cat_kernel<<<4096, 256, 0, stream>>>(agg, cat, l);
    }

    pair_kernel<<<(N_PAIRS + 7) / 8, 256, 0, stream>>>(
        uidx, iidx, cat, lin1w, lin1b, (float*)d_out);
}